// RGCN_49082886259001
// MI455X (gfx1250) — compile-verified
//
#include <hip/hip_runtime.h>

typedef __attribute__((ext_vector_type(2))) float v2f;
typedef __attribute__((ext_vector_type(8))) float v8f;

#define N_NODES 50000
#define N_REL 8
#define HIDDEN 128
#define N_EDGES 800000

#define KC      128              // K-chunk staged in LDS
#define NCHUNKS 9                // 8 chunks of W2 (1024) + 1 chunk of root2 (128)
#define LDSPAIR 288              // dword stride per K-pair row: 288 % 64 == 32
                                 //  -> the two WMMA half-waves hit disjoint bank groups

// ---------------------------------------------------------------------------
// Kernel 1: per-(dst, relation) edge counts for mean aggregation
// ---------------------------------------------------------------------------
__global__ void rgcn_count(const int* __restrict__ ei, const int* __restrict__ et,
                           float* __restrict__ cnt) {
    int e = blockIdx.x * blockDim.x + threadIdx.x;
    if (e >= N_EDGES) return;
    int d = ei[N_EDGES + e];
    int r = et[e];
    atomicAdd(&cnt[d * N_REL + r], 1.0f);
}

// ---------------------------------------------------------------------------
// Kernel 2: layer-1 message scatter: h[dst] += W1[rel, src] * norm
// one wave32 per edge; each lane owns 4 consecutive floats (32*4 = 128 = H)
// ---------------------------------------------------------------------------
__global__ void rgcn_scatter1(const int* __restrict__ ei, const int* __restrict__ et,
                              const float* __restrict__ W1, const float* __restrict__ cnt,
                              float* __restrict__ h) {
    int e = (blockIdx.x * blockDim.x + threadIdx.x) >> 5;
    int lane = threadIdx.x & 31;
    if (e >= N_EDGES) return;
    int s = ei[e];
    int d = ei[N_EDGES + e];
    int r = et[e];
    float norm = 1.0f / cnt[d * N_REL + r];
    const float4* wrow = (const float4*)(W1 + ((size_t)r * N_NODES + s) * HIDDEN);
    float4 v = wrow[lane];                       // coalesced 128-bit load
    float* hd = h + (size_t)d * HIDDEN + lane * 4;
    atomicAdd(hd + 0, v.x * norm);
    atomicAdd(hd + 1, v.y * norm);
    atomicAdd(hd + 2, v.z * norm);
    atomicAdd(hd + 3, v.w * norm);
}

// ---------------------------------------------------------------------------
// Kernel 3: h = elu(h + root1 + bias1)
// ---------------------------------------------------------------------------
__global__ void rgcn_finalize_h(float* __restrict__ h, const float* __restrict__ root1,
                                const float* __restrict__ bias1) {
    size_t i = (size_t)blockIdx.x * blockDim.x + threadIdx.x;
    if (i >= (size_t)N_NODES * HIDDEN) return;
    float x = h[i] + root1[i] + bias1[i & (HIDDEN - 1)];
    h[i] = x > 0.0f ? x : (__expf(x) - 1.0f);
}

// ---------------------------------------------------------------------------
// Kernel 4: layer-2 aggregation: agg[dst, rel] += h[src] * norm
// ---------------------------------------------------------------------------
__global__ void rgcn_scatter2(const int* __restrict__ ei, const int* __restrict__ et,
                              const float* __restrict__ h, const float* __restrict__ cnt,
                              float* __restrict__ agg) {
    int e = (blockIdx.x * blockDim.x + threadIdx.x) >> 5;
    int lane = threadIdx.x & 31;
    if (e >= N_EDGES) return;
    int s = ei[e];
    int d = ei[N_EDGES + e];
    int r = et[e];
    float norm = 1.0f / cnt[d * N_REL + r];
    const float4* hrow = (const float4*)(h + (size_t)s * HIDDEN);
    float4 v = hrow[lane];
    float* ad = agg + ((size_t)d * N_REL + r) * HIDDEN + lane * 4;
    atomicAdd(ad + 0, v.x * norm);
    atomicAdd(ad + 1, v.y * norm);
    atomicAdd(ad + 2, v.z * norm);
    atomicAdd(ad + 3, v.w * norm);
}

// ---------------------------------------------------------------------------
// Kernel 5: fused GEMM  out = elu([agg | h] · [W2 | root2] + bias2)
//   A = [50000 x 1152]  (agg reshaped [N,1024] ++ h [N,128])
//   B = [1152 x 128]    (W2 reshaped [1024,128] ++ root2 [128,128])
// Block = 8 waves (128 rows). B staged chunk-by-chunk into LDS in K-pair
// interleaved layout: ldsB[kp*LDSPAIR + col*2 + (k&1)] = B[2kp + (k&1)][col],
// so each lane's WMMA B-fragment {B[kr,col], B[kr+1,col]} is ONE ds_load_b64
// into an even-aligned VGPR pair (no register shuffles, conflict-free banks).
// One wave = 16-row x 128-col stripe: 8 f32 accumulators,
// V_WMMA_F32_16X16X4_F32, A fragment reused across all 8 column tiles.
// ---------------------------------------------------------------------------
__global__ void __launch_bounds__(256) rgcn_gemm(const float* __restrict__ agg,
                                                 const float* __restrict__ hbuf,
                                                 const float* __restrict__ W2,
                                                 const float* __restrict__ root2,
                                                 const float* __restrict__ bias2,
                                                 float* __restrict__ out) {
    __shared__ float ldsB[(KC / 2) * LDSPAIR];   // 64 * 288 * 4B = 72 KB

    const int nTiles = N_NODES / 16;             // 3125
    int tid  = threadIdx.x;
    int lane = tid & 31;
    int tile = blockIdx.x * 8 + (tid >> 5);      // one 16-row M tile per wave
    bool valid = tile < nTiles;                  // wave-uniform
    int hsel = lane >> 4;                        // 0: K {k,k+1}; 1: K {k+2,k+3}
    int l16  = lane & 15;
    int mrow = tile * 16 + l16;                  // A row this lane feeds

    v8f acc[8] = {};                             // 8 col-tiles x 8 VGPRs of C

    for (int chunk = 0; chunk < NCHUNKS; ++chunk) {
        // ---- stage B chunk [KC x 128] into pair-interleaved LDS ----
        const float* bsrc = (chunk < 8) ? (W2 + (size_t)chunk * KC * HIDDEN) : root2;
#pragma unroll
        for (int i = 0; i < 8; ++i) {
            int u  = tid + i * 256;              // staging unit 0..2047
            int kp = u >> 5;                     // pair-row 0..63
            int c  = (u & 31) * 4;               // col 0..124
            float4 g0 = *(const float4*)(bsrc + (size_t)(2 * kp) * HIDDEN + c);
            float4 g1 = *(const float4*)(bsrc + (size_t)(2 * kp + 1) * HIDDEN + c);
            float4 p0 = {g0.x, g1.x, g0.y, g1.y};    // cols c, c+1 interleaved
            float4 p1 = {g0.z, g1.z, g0.w, g1.w};    // cols c+2, c+3 interleaved
            *(float4*)&ldsB[kp * LDSPAIR + 2 * c]     = p0;
            *(float4*)&ldsB[kp * LDSPAIR + 2 * c + 4] = p1;
        }
        __syncthreads();

        if (valid) {
            const float* Arow = (chunk < 8)
                ? (agg + (size_t)mrow * (N_REL * HIDDEN) + chunk * KC)
                : (hbuf + (size_t)mrow * HIDDEN);
            for (int k = 0; k < KC; k += 4) {
                int kr = k + hsel * 2;
                v2f a = *(const v2f*)(Arow + kr);    // A[m, kr], A[m, kr+1]
                int kp = (k >> 1) + hsel;            // this lane's pair-row
#pragma unroll
                for (int t = 0; t < 8; ++t) {
                    int col = t * 16 + l16;
                    // {B[kr,col], B[kr+1,col]} in one 64-bit LDS read
                    v2f b = *(const v2f*)&ldsB[kp * LDSPAIR + col * 2];
                    acc[t] = __builtin_amdgcn_wmma_f32_16x16x4_f32(
                        false, a, false, b, (short)0, acc[t], false, false);
                }
            }
        }
        __syncthreads();
    }

    // ---- epilogue: bias + ELU + store (C/D layout: VGPR j -> row j / j+8) ----
    if (valid) {
#pragma unroll
        for (int t = 0; t < 8; ++t) {
            int col = t * 16 + l16;
            float bz = bias2[col];
#pragma unroll
            for (int j = 0; j < 8; ++j) {
                int row = tile * 16 + hsel * 8 + j;
                float v = acc[t][j] + bz;
                v = v > 0.0f ? v : (__expf(v) - 1.0f);
                out[(size_t)row * HIDDEN + col] = v;
            }
        }
    }
}

// ---------------------------------------------------------------------------
// Launcher
// ---------------------------------------------------------------------------
extern "C" void kernel_launch(void* const* d_in, const int* in_sizes, int n_in,
                              void* d_out, int out_size, void* d_ws, size_t ws_size,
                              hipStream_t stream) {
    const int*   edge_index = (const int*)d_in[0];   // [2, E]
    const int*   edge_type  = (const int*)d_in[1];   // [E]
    const float* W1         = (const float*)d_in[2]; // [R, N, H]
    const float* root1      = (const float*)d_in[3]; // [N, H]
    const float* bias1      = (const float*)d_in[4]; // [H]
    const float* W2         = (const float*)d_in[5]; // [R, H, H]
    const float* root2      = (const float*)d_in[6]; // [H, H]
    const float* bias2      = (const float*)d_in[7]; // [H]
    float*       out        = (float*)d_out;         // [N, H]

    // workspace layout: cnt [N*R] | h [N*H] | agg [N*R*H]   (~232 MB)
    float* cnt  = (float*)d_ws;
    float* hbuf = cnt + (size_t)N_NODES * N_REL;
    float* agg  = hbuf + (size_t)N_NODES * HIDDEN;
    size_t zero_bytes = ((size_t)N_NODES * N_REL +
                         (size_t)N_NODES * HIDDEN +
                         (size_t)N_NODES * N_REL * HIDDEN) * sizeof(float);
    hipMemsetAsync(d_ws, 0, zero_bytes, stream);  // capture-safe; required per call

    rgcn_count    <<<(N_EDGES + 255) / 256, 256, 0, stream>>>(edge_index, edge_type, cnt);
    rgcn_scatter1 <<<(N_EDGES + 7) / 8,     256, 0, stream>>>(edge_index, edge_type, W1, cnt, hbuf);
    rgcn_finalize_h<<<((size_t)N_NODES * HIDDEN + 255) / 256, 256, 0, stream>>>(hbuf, root1, bias1);
    rgcn_scatter2 <<<(N_EDGES + 7) / 8,     256, 0, stream>>>(edge_index, edge_type, hbuf, cnt, agg);

    int nTiles = N_NODES / 16;  // 3125
    rgcn_gemm<<<(nTiles + 7) / 8, 256, 0, stream>>>(agg, hbuf, W2, root2, bias2, out);
}